// StructuralProbe_71897752535526
// MI455X (gfx1250) — compile-verified
//
#include <hip/hip_runtime.h>

typedef __attribute__((ext_vector_type(2))) float v2f;
typedef __attribute__((ext_vector_type(8))) float v8f;

// Problem dims (fixed by the reference)
#define B_   32
#define L_   512
#define D_   1024
#define R_   128
#define ROWS (B_ * L_)          // 16384

// ---------------------------------------------------------------------------
// Kernel 1: T = X(16384 x 1024) @ P(1024 x 128), exact f32 via WMMA 16x16x4.
// Register-blocked: one wave computes a 64x32 tile of T (8 accumulators).
// Per K-step of 4: 4 A b64 loads + 4 B b32 loads feed 8 WMMAs (1 load/wmma).
// A layout (16x4 f32): lane<16 -> {K=k,K=k+1}, lane>=16 -> {K=k+2,K=k+3}
// B layout (4x16 f32): same K split across lane halves, N = lane&15.
// ---------------------------------------------------------------------------
__global__ __launch_bounds__(256) void gemm_xp(const float* __restrict__ X,
                                               const float* __restrict__ P,
                                               float* __restrict__ T) {
  const int wave = threadIdx.x >> 5;
  const int lane = threadIdx.x & 31;
  const int g    = blockIdx.x * 8 + wave;   // 0 .. 1023
  const int m0   = (g >> 2) * 64;           // 256 M-strips of 64
  const int n0   = (g & 3) * 32;            // 4 N-strips of 32
  const int half = lane >> 4;               // K sub-group (0/1)
  const int l16  = lane & 15;

  const float* a0 = X + (size_t)(m0 + l16) * D_ + 2 * half;
  const size_t aStride = (size_t)16 * D_;
  const float* b0 = P + (size_t)(2 * half) * R_ + n0 + l16;

  v8f c[4][2] = {};
  for (int k = 0; k < D_; k += 4) {
    v2f a[4];
#pragma unroll
    for (int i = 0; i < 4; ++i)
      a[i] = *(const v2f*)(a0 + aStride * i + k);   // X[m0+16i+l16][k+2h..+1]
    v2f bb[2];
#pragma unroll
    for (int j = 0; j < 2; ++j) {
      bb[j].x = b0[(size_t)k * R_ + 16 * j];        // P[k+2h  ][n0+16j+l16]
      bb[j].y = b0[(size_t)k * R_ + R_ + 16 * j];   // P[k+2h+1][n0+16j+l16]
    }
#pragma unroll
    for (int i = 0; i < 4; ++i)
#pragma unroll
      for (int j = 0; j < 2; ++j)
        c[i][j] = __builtin_amdgcn_wmma_f32_16x16x4_f32(
            false, a[i], false, bb[j], (short)0, c[i][j], false, false);
  }

  // C/D layout: lane l, VGPR v -> M = v + 8*(l>=16), N = l&15
#pragma unroll
  for (int i = 0; i < 4; ++i)
#pragma unroll
    for (int j = 0; j < 2; ++j)
#pragma unroll
      for (int v = 0; v < 8; ++v)
        T[(size_t)(m0 + 16 * i + v + 8 * half) * R_ + n0 + 16 * j + l16] =
            c[i][j][v];
}

// ---------------------------------------------------------------------------
// Kernel 2: sq_norms[r] = sum_k T[r][k]^2   (one thread per row, float4 reads)
// ---------------------------------------------------------------------------
__global__ __launch_bounds__(256) void row_sqnorm(const float* __restrict__ T,
                                                  float* __restrict__ N) {
  const int r = blockIdx.x * 256 + threadIdx.x;   // exactly 16384 threads
  const float4* p = (const float4*)(T + (size_t)r * R_);
  float s = 0.f;
#pragma unroll
  for (int i = 0; i < R_ / 4; ++i) {
    float4 v = p[i];
    s += v.x * v.x + v.y * v.y + v.z * v.z + v.w * v.w;
  }
  N[r] = s;
}

// ---------------------------------------------------------------------------
// Kernel 3: per batch b, C = T_b @ T_b^T (512x128 * 128x512), then
//           out = max(0, n_i + n_j - 2*C).
// Register-blocked: one wave computes a 32x32 output tile (4 accumulators).
// Per K-step of 4: 2 A b64 + 2 B b64 loads feed 4 WMMAs (1 b64/wmma).
// B operand is T_b^T, so B[k][n] = T_b[j0+n][k] -> contiguous float2 per lane.
// ---------------------------------------------------------------------------
__global__ __launch_bounds__(256) void pairdist(const float* __restrict__ T,
                                                const float* __restrict__ N,
                                                float* __restrict__ Out) {
  const int wave = threadIdx.x >> 5;
  const int lane = threadIdx.x & 31;
  const int g    = blockIdx.x * 8 + wave;   // 0 .. 8191
  const int b    = g >> 8;                  // 0 .. 31
  const int rem  = g & 255;
  const int i0   = (rem >> 4) * 32;         // 16 I-strips of 32
  const int j0   = (rem & 15) * 32;         // 16 J-strips of 32
  const int half = lane >> 4;
  const int l16  = lane & 15;

  const float* Tb = T + (size_t)b * L_ * R_;
  const float* Ar = Tb + (size_t)(i0 + l16) * R_ + 2 * half;
  const float* Br = Tb + (size_t)(j0 + l16) * R_ + 2 * half;
  const size_t rs = (size_t)16 * R_;

  v8f c[2][2] = {};
  for (int k = 0; k < R_; k += 4) {
    v2f a[2], bb[2];
    a[0]  = *(const v2f*)(Ar + k);
    a[1]  = *(const v2f*)(Ar + rs + k);
    bb[0] = *(const v2f*)(Br + k);
    bb[1] = *(const v2f*)(Br + rs + k);
#pragma unroll
    for (int i = 0; i < 2; ++i)
#pragma unroll
      for (int j = 0; j < 2; ++j)
        c[i][j] = __builtin_amdgcn_wmma_f32_16x16x4_f32(
            false, a[i], false, bb[j], (short)0, c[i][j], false, false);
  }

  const float* Nb = N + (size_t)b * L_;
  const float nj0 = Nb[j0 + l16];
  const float nj1 = Nb[j0 + 16 + l16];
  float* Ob = Out + (size_t)b * L_ * L_;
#pragma unroll
  for (int i = 0; i < 2; ++i)
#pragma unroll
    for (int v = 0; v < 8; ++v) {
      const int m = i0 + 16 * i + v + 8 * half;
      const float ni = Nb[m];
      float d0 = ni + nj0 - 2.0f * c[i][0][v];
      float d1 = ni + nj1 - 2.0f * c[i][1][v];
      Ob[(size_t)m * L_ + j0 + l16]      = d0 > 0.f ? d0 : 0.f;
      Ob[(size_t)m * L_ + j0 + 16 + l16] = d1 > 0.f ? d1 : 0.f;
    }
}

// ---------------------------------------------------------------------------
extern "C" void kernel_launch(void* const* d_in, const int* in_sizes, int n_in,
                              void* d_out, int out_size, void* d_ws, size_t ws_size,
                              hipStream_t stream) {
  (void)in_sizes; (void)n_in; (void)out_size; (void)ws_size;
  const float* X = (const float*)d_in[0];   // batch (32,512,1024)
  const float* P = (const float*)d_in[1];   // proj  (1024,128)
  float* Out = (float*)d_out;               // (32,512,512)

  float* T  = (float*)d_ws;                 // 16384*128 f32 = 8 MB
  float* Nn = T + (size_t)ROWS * R_;        // 16384 f32

  // waves: (ROWS/64)*(R_/32) = 1024 -> 128 blocks of 8 waves
  gemm_xp   <<<dim3(128),  dim3(256), 0, stream>>>(X, P, T);
  row_sqnorm<<<dim3(ROWS / 256), dim3(256), 0, stream>>>(T, Nn);
  // waves: B_*(L_/32)*(L_/32) = 8192 -> 1024 blocks of 8 waves
  pairdist  <<<dim3(1024), dim3(256), 0, stream>>>(T, Nn, Out);
}